// GConvLSTM_seq2seq_6794638262636
// MI455X (gfx1250) — compile-verified
//
#include <hip/hip_runtime.h>
#include <math.h>

// Problem constants (from reference)
#define NN   50000
#define EE   800000
#define TT   12
#define CIN  32
#define HH   64
#define OO   32

typedef __attribute__((ext_vector_type(2))) float v2f;
typedef __attribute__((ext_vector_type(8))) float v8f;

// ---------------- utility kernels ----------------

__global__ void zero_kernel(float* __restrict__ p, int n) {
    int i = blockIdx.x * blockDim.x + threadIdx.x;
    if (i < n) p[i] = 0.0f;
}

__global__ void deg_accum_kernel(const int* __restrict__ col,
                                 const float* __restrict__ attr,
                                 float* __restrict__ deg, int E) {
    int e = blockIdx.x * blockDim.x + threadIdx.x;
    if (e < E) atomicAdd(&deg[col[e]], attr[e]);
}

// deg -> dis = rsqrt(deg+1) (in place), self_norm = 1/(deg+1)
__global__ void node_norm_kernel(float* __restrict__ deg_dis,
                                 float* __restrict__ self_norm, int n) {
    int v = blockIdx.x * blockDim.x + threadIdx.x;
    if (v < n) {
        float d = deg_dis[v] + 1.0f;
        deg_dis[v]   = rsqrtf(d);
        self_norm[v] = 1.0f / d;
    }
}

__global__ void edge_norm_kernel(const int* __restrict__ row,
                                 const int* __restrict__ col,
                                 const float* __restrict__ attr,
                                 const float* __restrict__ dis,
                                 float* __restrict__ norm, int E) {
    int e = blockIdx.x * blockDim.x + threadIdx.x;
    if (e < E) norm[e] = dis[row[e]] * attr[e] * dis[col[e]];
}

// ---------------- WMMA fp32 GEMM ----------------
// out[NN x C] = concat(x1[NN x K1], x2[NN x K2]) @ W^T, W stored [C, K1+K2] row-major.
// Block = 16-row stripe, C/16 waves; shared LDS A-tile (concat resolved once,
// coalesced); per-wave 16x16 tile via fully-unrolled V_WMMA_F32_16X16X4_F32 chain.
// All dims exact multiples of 16/4 -> no predication, EXEC all-1s as WMMA requires.
template <int K1, int K2, int C>
__global__ __launch_bounds__(2 * C)
void gemm_wmma_kernel(const float* __restrict__ x1,
                      const float* __restrict__ x2,
                      const float* __restrict__ W,
                      float* __restrict__ out) {
    constexpr int K       = K1 + K2;
    constexpr int LDK     = K + 4;       // pad: gcd(K+4,64)=4 -> 16 distinct banks
    constexpr int THREADS = 2 * C;       // 32 * (C/16)
    __shared__ float tile[16 * LDK];

    const int tid = threadIdx.x;
    const int M0  = blockIdx.x * 16;

    // Cooperative, coalesced load of A-tile (rows M0..M0+15 of [x1|x2])
#pragma unroll
    for (int i = tid; i < 16 * K; i += THREADS) {
        const int r = i / K;
        const int k = i % K;
        const int m = M0 + r;
        const float v = (k < K1) ? x1[(size_t)m * K1 + k]
                                 : x2[(size_t)m * K2 + (k - K1)];
        tile[r * LDK + k] = v;
    }
    __syncthreads();

    const int lane = tid & 31;
    const int ml   = lane & 15;
    const int hi   = lane >> 4;          // selects K pair within each 4-step
    const int N0   = (tid >> 5) * 16;    // one 16-col tile per wave
    const float* wrow = W + (size_t)(N0 + ml) * K;
    const float* arow = tile + ml * LDK + 2 * hi;

    v8f acc = {};
#pragma unroll
    for (int kk = 0; kk < K; kk += 4) {
        v2f a = *(const v2f*)(arow + kk);        // ds_load_b64, branch-free
        v2f b = *(const v2f*)(wrow + kk + 2 * hi); // 8B-aligned (K even, offsets even)
        acc = __builtin_amdgcn_wmma_f32_16x16x4_f32(
                  false, a, false, b, (short)0, acc, false, false);
    }

#pragma unroll
    for (int r = 0; r < 8; ++r) {
        const int mr = M0 + r + 8 * hi;  // C/D layout: VGPR r = M=r / M=r+8
        out[(size_t)mr * C + N0 + ml] = acc[r];
    }
}

// ---------------- aggregation ----------------

// agg[v,f] = self_norm[v]*xlin[v,f] + b[f], float4-vectorized (F % 4 == 0)
__global__ void init_agg_kernel(const float4* __restrict__ xlin,
                                const float* __restrict__ self_norm,
                                const float4* __restrict__ bias,
                                float4* __restrict__ agg, int F4, int n) {
    size_t i = (size_t)blockIdx.x * blockDim.x + threadIdx.x;
    if (i < (size_t)n * F4) {
        int v = (int)(i / F4);
        int f = (int)(i % F4);
        float s = self_norm[v];
        float4 xv = xlin[i];
        float4 bv = bias[f];
        float4 o;
        o.x = s * xv.x + bv.x; o.y = s * xv.y + bv.y;
        o.z = s * xv.z + bv.z; o.w = s * xv.w + bv.w;
        agg[i] = o;
    }
}

// wave per edge: agg[col,:] += norm_e * xlin[row,:]  (atomics resolve in L2)
__global__ void edge_agg_kernel(const int* __restrict__ row,
                                const int* __restrict__ col,
                                const float* __restrict__ norm,
                                const float* __restrict__ xlin,
                                float* __restrict__ agg, int F, int E) {
    int wid  = (blockIdx.x * blockDim.x + threadIdx.x) >> 5;
    int lane = threadIdx.x & 31;
    if (wid >= E) return;
    int r = row[wid], c = col[wid];
    float nv = norm[wid];
    const float* src = xlin + (size_t)r * F;
    float*       dst = agg  + (size_t)c * F;
    if (F >= 128) {
        // float4 gather + 4 scalar atomics per chunk
        for (int f = lane * 4; f < F; f += 128) {
            float4 v = *(const float4*)(src + f);
            atomicAdd(dst + f + 0, nv * v.x);
            atomicAdd(dst + f + 1, nv * v.y);
            atomicAdd(dst + f + 2, nv * v.z);
            atomicAdd(dst + f + 3, nv * v.w);
        }
    } else {
        for (int f = lane; f < F; f += 32)
            atomicAdd(dst + f, nv * src[f]);
    }
}

// ---------------- LSTM pointwise ----------------
// cc = [n, 4*Hc] gates (i,f,o,g); updates h,c in place; optionally emits h2.
__global__ void lstm_kernel(const float* __restrict__ cc,
                            float* __restrict__ h, float* __restrict__ c,
                            int Hc, int n, float* __restrict__ out_t) {
    size_t i = (size_t)blockIdx.x * blockDim.x + threadIdx.x;
    if (i >= (size_t)n * Hc) return;
    int v = (int)(i / Hc);
    int j = (int)(i % Hc);
    const float* g = cc + (size_t)v * 4 * Hc;
    float ig = g[j], fg = g[Hc + j], og = g[2 * Hc + j], gg = g[3 * Hc + j];
    float si = 1.0f / (1.0f + expf(-ig));
    float sf = 1.0f / (1.0f + expf(-fg));
    float so = 1.0f / (1.0f + expf(-og));
    float c2 = sf * c[i] + si * tanhf(gg);
    float h2 = so * tanhf(c2);
    c[i] = c2;
    h[i] = h2;
    if (out_t) out_t[i] = h2;
}

// st[n,64] -> hdec = elu(st[:, :32]), cdec = elu(st[:, 32:])
__global__ void elu_split_kernel(const float* __restrict__ st,
                                 float* __restrict__ hdec,
                                 float* __restrict__ cdec, int n) {
    size_t i = (size_t)blockIdx.x * blockDim.x + threadIdx.x;
    if (i >= (size_t)n * OO) return;
    int v = (int)(i / OO);
    int j = (int)(i % OO);
    float a = st[(size_t)v * 64 + j];
    float b = st[(size_t)v * 64 + 32 + j];
    hdec[i] = (a > 0.0f) ? a : (expf(a) - 1.0f);
    cdec[i] = (b > 0.0f) ? b : (expf(b) - 1.0f);
}

// ---------------- host orchestration ----------------

static inline int ceil_div(int a, int b) { return (a + b - 1) / b; }

extern "C" void kernel_launch(void* const* d_in, const int* in_sizes, int n_in,
                              void* d_out, int out_size, void* d_ws, size_t ws_size,
                              hipStream_t stream) {
    const float* x          = (const float*)d_in[0];   // [T, N, CIN]
    const int*   edge_index = (const int*)d_in[1];     // [2, E]
    const float* edge_attr  = (const float*)d_in[2];   // [E]
    // d_in[3] = seq_len (fixed 12)
    const float* W_enc      = (const float*)d_in[4];   // [256, 96]
    const float* b_enc      = (const float*)d_in[5];   // [256]
    const float* W_dec_init = (const float*)d_in[6];   // [64, 128]
    const float* b_dec_init = (const float*)d_in[7];   // [64]
    const float* W_dec_cell = (const float*)d_in[8];   // [128, 96]
    const float* b_dec_cell = (const float*)d_in[9];   // [128]
    float* out = (float*)d_out;                         // [12, N, 32]

    const int* row = edge_index;
    const int* col = edge_index + EE;

    // workspace layout (floats); all region sizes are multiples of 4 -> 16B aligned
    float* ws   = (float*)d_ws;
    float* dis  = ws;            ws += NN;              // deg -> rsqrt(deg+1)
    float* sn   = ws;            ws += NN;              // self_norm
    float* norm = ws;            ws += EE;
    float* h    = ws;            ws += (size_t)NN * HH;
    float* c    = ws;            ws += (size_t)NN * HH;
    float* hdec = ws;            ws += (size_t)NN * OO;
    float* cdec = ws;            ws += (size_t)NN * OO;
    float* xlin = ws;            ws += (size_t)NN * 256;
    float* agg  = ws;            ws += (size_t)NN * 256;

    const int B = 256;

    // --- normalization ---
    zero_kernel<<<ceil_div(NN, B), B, 0, stream>>>(dis, NN);
    deg_accum_kernel<<<ceil_div(EE, B), B, 0, stream>>>(col, edge_attr, dis, EE);
    node_norm_kernel<<<ceil_div(NN, B), B, 0, stream>>>(dis, sn, NN);
    edge_norm_kernel<<<ceil_div(EE, B), B, 0, stream>>>(row, col, edge_attr, dis, norm, EE);

    // --- init encoder state ---
    zero_kernel<<<ceil_div(NN * HH, B), B, 0, stream>>>(h, NN * HH);
    zero_kernel<<<ceil_div(NN * HH, B), B, 0, stream>>>(c, NN * HH);

    const int edgeWaves = ceil_div(EE * 32, B);
    const int rowTiles  = NN / 16;

    // --- encoder: 12 steps, C = 4*H = 256, K = 32+64 ---
    for (int t = 0; t < TT; ++t) {
        const float* xt = x + (size_t)t * NN * CIN;
        gemm_wmma_kernel<CIN, HH, 256><<<rowTiles, 512, 0, stream>>>(
            xt, h, W_enc, xlin);
        init_agg_kernel<<<ceil_div(NN * 64, B), B, 0, stream>>>(
            (const float4*)xlin, sn, (const float4*)b_enc, (float4*)agg, 64, NN);
        edge_agg_kernel<<<edgeWaves, B, 0, stream>>>(
            row, col, norm, xlin, agg, 256, EE);
        lstm_kernel<<<ceil_div(NN * HH, B), B, 0, stream>>>(
            agg, h, c, HH, NN, nullptr);
    }

    // --- decoder init: C = 2*O = 64, K = 64+64 ---
    gemm_wmma_kernel<HH, HH, 64><<<rowTiles, 128, 0, stream>>>(
        h, c, W_dec_init, xlin);
    init_agg_kernel<<<ceil_div(NN * 16, B), B, 0, stream>>>(
        (const float4*)xlin, sn, (const float4*)b_dec_init, (float4*)agg, 16, NN);
    edge_agg_kernel<<<edgeWaves, B, 0, stream>>>(
        row, col, norm, xlin, agg, 64, EE);
    elu_split_kernel<<<ceil_div(NN * OO, B), B, 0, stream>>>(agg, hdec, cdec, NN);

    // --- decoder: 12 steps, input = encoder h (constant), C = 4*O = 128, K = 64+32 ---
    for (int t = 0; t < TT; ++t) {
        gemm_wmma_kernel<HH, OO, 128><<<rowTiles, 256, 0, stream>>>(
            h, hdec, W_dec_cell, xlin);
        init_agg_kernel<<<ceil_div(NN * 32, B), B, 0, stream>>>(
            (const float4*)xlin, sn, (const float4*)b_dec_cell, (float4*)agg, 32, NN);
        edge_agg_kernel<<<edgeWaves, B, 0, stream>>>(
            row, col, norm, xlin, agg, 128, EE);
        lstm_kernel<<<ceil_div(NN * OO, B), B, 0, stream>>>(
            agg, hdec, cdec, OO, NN, out + (size_t)t * NN * OO);
    }
}